// Operator2DBase_1099511628448
// MI455X (gfx1250) — compile-verified
//
#include <hip/hip_runtime.h>
#include <stdint.h>

// ---------------------------------------------------------------------------
// Per-pixel (unshared) 5x5 convolution:
//   out[b,y,x] = relu( f[b,y,x] + dt[b] * sum_{i,j} f_pad[b,y+i,x+j]*K[i,j,y,x] )
//
// Roofline: ~0.45 GFLOP vs ~90 MB compulsory traffic (~5 FLOP/byte) -> purely
// HBM-bound on MI455X (23.3 TB/s => ~4 us floor). WMMA is structurally
// inapplicable (both contraction operands are pixel-dependent), so the CDNA5
// win is data movement:
//  * block = 32x8 pixel tile, loops over all 32 batches: each per-pixel K tap
//    is read from HBM once into VGPRs and reused 32x (K traffic stays 26 MB).
//  * interior blocks: f tile (36x12 incl. halo) staged by the Tensor Data
//    Mover -- ONE tensor_load_to_lds (2D D#) per batch, double-buffered on
//    TENSORcnt so DMA of batch b+1 overlaps compute of batch b.
//  * boundary blocks: per-lane global_load_async_to_lds_b32 with clamped
//    addresses + per-tap zero-padding masks (ASYNCcnt double buffer).
// ---------------------------------------------------------------------------

namespace {

constexpr int H  = 512, W = 512, B = 32;
constexpr int KS = 5, R = 2;              // kernel size / radius (pl = pr = 2)
constexpr int TW = 32, TH = 8;            // output tile per block
constexpr int PW = TW + KS - 1;           // 36  padded tile width
constexpr int PH = TH + KS - 1;           // 12  padded tile height
constexpr int PTOT = PW * PH;             // 432 padded elements
constexpr int NTHREADS = TW * TH;         // 256 threads = 8 waves (wave32)

static_assert(TW == 32, "ty = t >> 5 assumes TW == 32");
static_assert(PTOT <= 2 * NTHREADS, "two async loads per thread cover the tile");

typedef unsigned u32x4 __attribute__((ext_vector_type(4)));
typedef unsigned u32x8 __attribute__((ext_vector_type(8)));

__device__ __forceinline__ void async_ld_f32(unsigned lds_byte_off, const float* g) {
  unsigned long long ga = (unsigned long long)(uintptr_t)g;
  // gfx1250 VGLOBAL async load, GV mode (vdst = LDS byte address, 64-bit vaddr)
  asm volatile("global_load_async_to_lds_b32 %0, %1, off"
               :: "v"(lds_byte_off), "v"(ga)
               : "memory");
}

template <int N>
__device__ __forceinline__ void wait_asynccnt() {
#if __has_builtin(__builtin_amdgcn_s_wait_asynccnt)
  __builtin_amdgcn_s_wait_asynccnt(N);
#else
  asm volatile("s_wait_asynccnt %0" :: "n"(N) : "memory");
#endif
}

template <int N>
__device__ __forceinline__ void wait_tensorcnt() {
#if __has_builtin(__builtin_amdgcn_s_wait_tensorcnt)
  __builtin_amdgcn_s_wait_tensorcnt(N);
#else
  asm volatile("s_wait_tensorcnt %0" :: "n"(N) : "memory");
#endif
}

// One wave-level TDM DMA: 2D tile PW x PH (f32) from f[b] starting at
// (y0-R, x0-R) into LDS at lds_byte. D# per CDNA5 ISA.
__device__ __forceinline__ void tdm_load_tile(const float* f, int b, int y0, int x0,
                                              unsigned lds_byte) {
  const uint64_t ga = (uint64_t)(uintptr_t)f +
      4ull * ((uint64_t)b * H * W + (uint64_t)(y0 - R) * W + (uint64_t)(x0 - R));
  u32x4 g0;
  g0[0] = 1u;                                              // count=1, user D#
  g0[1] = lds_byte;                                        // lds_addr[31:0]
  g0[2] = (unsigned)ga;                                    // global_addr[31:0]
  g0[3] = (unsigned)((ga >> 32) & 0x01FFFFFFull)           // global_addr[56:32]
          | 0x80000000u;                                   // type = 2 ("image")

  u32x8 g1;
  g1[0] = 0x00020000u;                                     // data_size = 4B; mask=0
  g1[1] = ((unsigned)W & 0xFFFFu) << 16;                   // tensor_dim0[15:0]
  g1[2] = ((unsigned)W >> 16)                              // tensor_dim0[31:16]
          | (((unsigned)H & 0xFFFFu) << 16);               // tensor_dim1[15:0]
  g1[3] = ((unsigned)H >> 16)                              // tensor_dim1[31:16]
          | ((unsigned)PW << 16);                          // tile_dim0 = 36
  g1[4] = (unsigned)PH;                                    // tile_dim1 = 12 (dim2=0)
  g1[5] = (unsigned)W;                                     // dim0_stride[31:0] = 512
  g1[6] = 0u;                                              // stride hi / dim1_stride lo
  g1[7] = 0u;
  // 2-group form: VADDR2/VADDR3 = NULL -> tensor up to 2D. TDM ignores EXEC,
  // so exactly one wave per block must execute this.
  asm volatile("tensor_load_to_lds %0, %1" :: "s"(g0), "s"(g1) : "memory");
}

__device__ __forceinline__ int clampi(int v, int lo, int hi) {
  return v < lo ? lo : (v > hi ? hi : v);
}

__global__ __launch_bounds__(NTHREADS)
void conv5x5_perpixel_kernel(const float* __restrict__ f,
                             const float* __restrict__ Kw,
                             const float* __restrict__ dt,
                             float* __restrict__ out) {
  __shared__ float tile[2][PTOT];

  const int t  = threadIdx.x;
  const int tx = t & (TW - 1);
  const int ty = t >> 5;
  const int x0 = blockIdx.x * TW;
  const int y0 = blockIdx.y * TH;
  const int x  = x0 + tx;
  const int y  = y0 + ty;

  // ---- K taps into registers: each K element is read exactly once per
  //      dispatch and reused for all 32 batches from VGPRs. ----
  float kr[KS * KS];
#pragma unroll
  for (int i = 0; i < KS; ++i)
#pragma unroll
    for (int j = 0; j < KS; ++j)
      kr[i * KS + j] = Kw[((size_t)(i * KS + j) * H + y) * W + x];

  const unsigned lds0 = (unsigned)(uintptr_t)&tile[0][0];  // LDS byte offset
  const bool blk_interior = (x0 >= R) && (x0 + TW + R <= W) &&
                            (y0 >= R) && (y0 + TH + R <= H);  // block-uniform

  if (blk_interior) {
    // ================= TDM path (868 / 1024 blocks) =================
    const bool issuer = (t < 32);  // wave 0 only: TDM ignores EXEC
    if (issuer) tdm_load_tile(f, 0, y0, x0, lds0);

    for (int b = 0; b < B; ++b) {
      const int cur = b & 1;
      if (issuer) {
        if (b + 1 < B) {  // DMA next batch, then wait for current (in-order)
          tdm_load_tile(f, b + 1, y0, x0, lds0 + 4u * (unsigned)((1 - cur) * PTOT));
          wait_tensorcnt<1>();
        } else {
          wait_tensorcnt<0>();
        }
      }
      __syncthreads();  // tile[cur] valid for all 8 waves

      const float* tb = tile[cur];
      float df = 0.0f;
#pragma unroll
      for (int i = 0; i < KS; ++i)
#pragma unroll
        for (int j = 0; j < KS; ++j)
          df = fmaf(tb[(ty + i) * PW + (tx + j)], kr[i * KS + j], df);

      const float fc = tb[(ty + R) * PW + (tx + R)];       // == f[b,y,x]
      const float o  = fmaf(df, dt[b], fc);
      out[((size_t)b * H + y) * W + x] = o > 0.0f ? o : 0.0f;

      __syncthreads();  // done reading tile[cur] before it is restaged
    }
  } else {
    // ============ boundary path: per-lane async loads + masks ============
    const int e0 = t;
    const int e1 = (t + NTHREADS < PTOT) ? (t + NTHREADS) : (PTOT - 1);
    const int off0 = clampi(y0 + (e0 / PW) - R, 0, H - 1) * W +
                     clampi(x0 + (e0 % PW) - R, 0, W - 1);
    const int off1 = clampi(y0 + (e1 / PW) - R, 0, H - 1) * W +
                     clampi(x0 + (e1 % PW) - R, 0, W - 1);

    unsigned rowm = 0, colm = 0;   // zero-padding tap validity
#pragma unroll
    for (int i = 0; i < KS; ++i) {
      if ((unsigned)(y + i - R) < (unsigned)H) rowm |= 1u << i;
      if ((unsigned)(x + i - R) < (unsigned)W) colm |= 1u << i;
    }

    async_ld_f32(lds0 + 4u * (unsigned)e0, f + off0);
    async_ld_f32(lds0 + 4u * (unsigned)e1, f + off1);

    for (int b = 0; b < B; ++b) {
      const int cur = b & 1;
      if (b + 1 < B) {
        const float* fb = f + (size_t)(b + 1) * H * W;
        const unsigned ldsn = lds0 + 4u * (unsigned)((1 - cur) * PTOT);
        async_ld_f32(ldsn + 4u * (unsigned)e0, fb + off0);
        async_ld_f32(ldsn + 4u * (unsigned)e1, fb + off1);
        wait_asynccnt<2>();   // ASYNCcnt completes in issue order
      } else {
        wait_asynccnt<0>();
      }
      __syncthreads();

      const float* tb = tile[cur];
      float df = 0.0f;
#pragma unroll
      for (int i = 0; i < KS; ++i)
#pragma unroll
        for (int j = 0; j < KS; ++j) {
          const bool ok = ((rowm >> i) & (colm >> j) & 1u) != 0u;
          const float v = ok ? tb[(ty + i) * PW + (tx + j)] : 0.0f;
          df = fmaf(v, kr[i * KS + j], df);
        }

      const float fc = tb[(ty + R) * PW + (tx + R)];
      const float o  = fmaf(df, dt[b], fc);
      out[((size_t)b * H + y) * W + x] = o > 0.0f ? o : 0.0f;

      __syncthreads();
    }
  }
}

}  // namespace

extern "C" void kernel_launch(void* const* d_in, const int* in_sizes, int n_in,
                              void* d_out, int out_size, void* d_ws, size_t ws_size,
                              hipStream_t stream) {
  const float* f  = (const float*)d_in[0];   // (B,H,W)   float32
  const float* Kw = (const float*)d_in[1];   // (5,5,H,W) float32
  const float* dt = (const float*)d_in[2];   // (B,)      float32
  float* out      = (float*)d_out;           // (B,H,W)   float32

  dim3 grid(W / TW, H / TH);                 // 16 x 64 blocks
  conv5x5_perpixel_kernel<<<grid, NTHREADS, 0, stream>>>(f, Kw, dt, out);
}